// GATEncoder_8521215115851
// MI455X (gfx1250) — compile-verified
//
#include <hip/hip_runtime.h>

// ---------------- problem constants (match reference) ----------------
#define N_NODES 50000
#define E_EDGES 800000
#define ETOT    (E_EDGES + N_NODES)   // self-loops appended
#define G_GRAPHS 128
#define IN_DIM  128
#define HD      128                   // H*D
#define NHEAD   4
#define DHEAD   32
#define NLAYERS 3
#define CONDF   7
#define LATD    32
#define FEATS   (HD + CONDF)          // 135
#define NEG_SLOPE 0.2f
#define LDS_PITCH 132                 // padded row stride: 132 mod 64 = 4 -> conflict-free

typedef __attribute__((ext_vector_type(2))) float v2f;
typedef __attribute__((ext_vector_type(4))) float v4f;
typedef __attribute__((ext_vector_type(8))) float v8f;

// ---------------- utility ----------------
__global__ void fill_kernel(float* __restrict__ p, float v, int n) {
    int t = blockIdx.x * blockDim.x + threadIdx.x;
    if (t < n) p[t] = v;
}

__device__ __forceinline__ void atomicMaxF(float* addr, float val) {
    unsigned int* ua = (unsigned int*)addr;
    unsigned int old = *ua;
    while (__uint_as_float(old) < val) {
        unsigned int assumed = old;
        old = atomicCAS(ua, assumed, __float_as_uint(val));
        if (old == assumed) break;
    }
}

__device__ __forceinline__ float leakyf(float x, float s) {
    return x > 0.f ? x : s * x;
}

// ---------------- dense GEMM: H = X @ W  via V_WMMA_F32_16X16X4_F32 ---------
// grid = 3125 blocks (16 rows each), 256 threads = 8 waves, wave w -> 16-col tile w.
// A tile (16x128, 8KB) staged once per block into LDS (shared by all 8 waves);
// W slices stream from global (L2-resident, no intra-block reuse).
__global__ __launch_bounds__(256) void gemm_wmma_kernel(
        const float* __restrict__ X, const float* __restrict__ Wm,
        float* __restrict__ Hout) {
    __shared__ float As[16 * LDS_PITCH];          // padded 16x128 A tile

    const int row0 = blockIdx.x * 16;

    // cooperative A-tile load: 16 threads per row, 8 floats per thread (2x b128)
    {
        const int r  = threadIdx.x >> 4;          // 0..15
        const int c0 = (threadIdx.x & 15) * 8;    // 0,8,...,120
        const float* src = X + (size_t)(row0 + r) * IN_DIM + c0;
        float* dst = As + r * LDS_PITCH + c0;
        *(v4f*)(dst)     = *(const v4f*)(src);
        *(v4f*)(dst + 4) = *(const v4f*)(src + 4);
    }
    __syncthreads();

    const int lane = threadIdx.x & 31;
    const int wave = threadIdx.x >> 5;            // 0..7 -> column tile
    const int half = lane >> 4;                   // 0 or 1
    const int l16  = lane & 15;
    const int col0 = wave * 16;

    v8f acc = {0.f, 0.f, 0.f, 0.f, 0.f, 0.f, 0.f, 0.f};
    const float* arow = As + l16 * LDS_PITCH;     // this lane's A row in LDS

#pragma unroll 4
    for (int kb = 0; kb < IN_DIM; kb += 4) {
        const int ka = kb + half * 2;
        v2f a = *(const v2f*)(arow + ka);         // ds_load_b64, conflict-free
        v2f b;
        b.x = Wm[(size_t)ka * HD + col0 + l16];
        b.y = Wm[(size_t)(ka + 1) * HD + col0 + l16];
        acc = __builtin_amdgcn_wmma_f32_16x16x4_f32(
                false, a, false, b, (short)0, acc, false, false);
    }
#pragma unroll
    for (int i = 0; i < 8; ++i) {
        const int m = i + half * 8;
        Hout[(size_t)(row0 + m) * HD + col0 + l16] = acc[i];
    }
}

// ---------------- attention logits: al_s/al_d [N,H] ----------------
__global__ void attn_scores_kernel(const float* __restrict__ Hbuf,
                                   const float* __restrict__ a_s,
                                   const float* __restrict__ a_d,
                                   float* __restrict__ al_s,
                                   float* __restrict__ al_d) {
    int t = blockIdx.x * blockDim.x + threadIdx.x;
    if (t >= N_NODES * NHEAD) return;
    const int n = t >> 2, h = t & 3;
    const float* hp = Hbuf + (size_t)n * HD + h * DHEAD;
    const float* as = a_s + h * DHEAD;
    const float* ad = a_d + h * DHEAD;
    float ss = 0.f, sd = 0.f;
#pragma unroll
    for (int d = 0; d < DHEAD; ++d) {
        float v = hp[d];
        ss += v * as[d];
        sd += v * ad[d];
    }
    al_s[t] = ss;
    al_d[t] = sd;
}

// ---------------- edge pass 1: e = leaky(als[src]+ald[dst]); segment max ----
__global__ void edge_max_kernel(const int* __restrict__ ei,
                                const float* __restrict__ al_s,
                                const float* __restrict__ al_d,
                                float* __restrict__ escr,
                                float* __restrict__ mmax) {
    int t = blockIdx.x * blockDim.x + threadIdx.x;
    if (t >= ETOT * NHEAD) return;
    const int e = t >> 2, h = t & 3;
    const int src = (e < E_EDGES) ? ei[e] : (e - E_EDGES);
    const int dst = (e < E_EDGES) ? ei[E_EDGES + e] : (e - E_EDGES);
    float v = leakyf(al_s[src * NHEAD + h] + al_d[dst * NHEAD + h], NEG_SLOPE);
    escr[t] = v;
    atomicMaxF(&mmax[dst * NHEAD + h], v);
}

// ---------------- edge pass 2: ex = exp(e - m[dst]); segment sum ------------
__global__ void edge_exp_kernel(const int* __restrict__ ei,
                                const float* __restrict__ mmax,
                                float* __restrict__ escr,
                                float* __restrict__ denom) {
    int t = blockIdx.x * blockDim.x + threadIdx.x;
    if (t >= ETOT * NHEAD) return;
    const int e = t >> 2, h = t & 3;
    const int dst = (e < E_EDGES) ? ei[E_EDGES + e] : (e - E_EDGES);
    float ex = __expf(escr[t] - mmax[dst * NHEAD + h]);
    escr[t] = ex;
    atomicAdd(&denom[dst * NHEAD + h], ex);
}

// ---------------- edge pass 3: out[dst] += alpha * h[src] -------------------
// one thread per (edge, 4-channel group): b128 gathers, coalesced f32 atomics.
__global__ void edge_agg_kernel(const int* __restrict__ ei,
                                const float* __restrict__ escr,
                                const float* __restrict__ denom,
                                const float* __restrict__ Hbuf,
                                float* __restrict__ Out) {
    int t = blockIdx.x * blockDim.x + threadIdx.x;
    if (t >= ETOT * (HD / 4)) return;
    const int q = t & (HD / 4 - 1);               // 0..31 -> channels 4q..4q+3
    const int e = t >> 5;
    const int h = q >> 3;                         // head = (4q)/32
    const int src = (e < E_EDGES) ? ei[e] : (e - E_EDGES);
    const int dst = (e < E_EDGES) ? ei[E_EDGES + e] : (e - E_EDGES);
    const float alpha = escr[e * NHEAD + h] / (denom[dst * NHEAD + h] + 1e-16f);
    const v4f hv = *(const v4f*)(Hbuf + (size_t)src * HD + q * 4);
    float* op = Out + (size_t)dst * HD + q * 4;
    atomicAdd(op + 0, alpha * hv.x);
    atomicAdd(op + 1, alpha * hv.y);
    atomicAdd(op + 2, alpha * hv.z);
    atomicAdd(op + 3, alpha * hv.w);
}

// ---------------- bias + LeakyReLU(0.2), writes next-layer input ------------
__global__ void bias_leaky_kernel(const float* __restrict__ In,
                                  const float* __restrict__ bias,
                                  float* __restrict__ Outp) {
    int t = blockIdx.x * blockDim.x + threadIdx.x;
    if (t >= N_NODES * HD) return;
    const int c = t & (HD - 1);
    Outp[t] = leakyf(In[t] + bias[c], 0.2f);
}

// ---------------- global add pool -------------------------------------------
__global__ void pool_kernel(const float* __restrict__ Xf,
                            const int* __restrict__ batch,
                            float* __restrict__ pooled) {
    int t = blockIdx.x * blockDim.x + threadIdx.x;
    if (t >= N_NODES * HD) return;
    const int c = t & (HD - 1);
    const int n = t >> 7;
    atomicAdd(&pooled[batch[n] * HD + c], Xf[t]);
}

// ---------------- BN stats over 128 graphs (biased var) ---------------------
__global__ __launch_bounds__(256) void bn_stats_kernel(
        const float* __restrict__ pooled, const float* __restrict__ cond,
        const float* __restrict__ gamma, const float* __restrict__ beta,
        float* __restrict__ scale, float* __restrict__ shift) {
    int k = threadIdx.x;
    if (k >= FEATS) return;
    float s = 0.f, s2 = 0.f;
    for (int g = 0; g < G_GRAPHS; ++g) {
        float v = (k < HD) ? pooled[g * HD + k] : cond[g * CONDF + (k - HD)];
        s += v;
        s2 += v * v;
    }
    const float mu  = s  * (1.0f / G_GRAPHS);
    const float var = s2 * (1.0f / G_GRAPHS) - mu * mu;
    const float sc  = gamma[k] * rsqrtf(var + 1e-5f);
    scale[k] = sc;
    shift[k] = beta[k] - mu * sc;
}

// ---------------- fused BN-apply + FC: out[g,l] ------------------------------
__global__ void fc_kernel(const float* __restrict__ pooled,
                          const float* __restrict__ cond,
                          const float* __restrict__ scale,
                          const float* __restrict__ shift,
                          const float* __restrict__ fcw,
                          const float* __restrict__ fcb,
                          float* __restrict__ out) {
    int t = blockIdx.x * blockDim.x + threadIdx.x;
    if (t >= G_GRAPHS * LATD) return;
    const int g = t / LATD, l = t % LATD;
    float acc = fcb[l];
    for (int k = 0; k < FEATS; ++k) {
        float v = (k < HD) ? pooled[g * HD + k] : cond[g * CONDF + (k - HD)];
        acc += (v * scale[k] + shift[k]) * fcw[k * LATD + l];
    }
    out[t] = acc;
}

// ---------------- host orchestration ----------------------------------------
extern "C" void kernel_launch(void* const* d_in, const int* in_sizes, int n_in,
                              void* d_out, int out_size, void* d_ws, size_t ws_size,
                              hipStream_t stream) {
    const float* x      = (const float*)d_in[0];
    const float* cond   = (const float*)d_in[1];
    const float* Wall   = (const float*)d_in[2];   // [L,128,128]
    const float* a_src  = (const float*)d_in[3];   // [L,4,32]
    const float* a_dst  = (const float*)d_in[4];
    const float* bias   = (const float*)d_in[5];   // [L,128]
    const float* gamma  = (const float*)d_in[6];
    const float* beta   = (const float*)d_in[7];
    const float* fcw    = (const float*)d_in[8];
    const float* fcb    = (const float*)d_in[9];
    const int*   ei     = (const int*)d_in[10];    // [2,E]
    const int*   batch  = (const int*)d_in[11];
    float*       out    = (float*)d_out;

    // workspace carve-out (floats)
    float* ws = (float*)d_ws;
    size_t off = 0;
    float* bufA   = ws + off; off += (size_t)N_NODES * HD;
    float* bufB   = ws + off; off += (size_t)N_NODES * HD;
    float* hbuf   = ws + off; off += (size_t)N_NODES * HD;
    float* als    = ws + off; off += (size_t)N_NODES * NHEAD;
    float* ald    = ws + off; off += (size_t)N_NODES * NHEAD;
    float* mmax   = ws + off; off += (size_t)N_NODES * NHEAD;
    float* denom  = ws + off; off += (size_t)N_NODES * NHEAD;
    float* escr   = ws + off; off += (size_t)ETOT * NHEAD;
    float* pooled = ws + off; off += (size_t)G_GRAPHS * HD;
    float* scale  = ws + off; off += 256;
    float* shift  = ws + off; off += 256;
    (void)ws_size; (void)in_sizes; (void)n_in; (void)out_size;

    const int B = 256;
    const int nNH  = N_NODES * NHEAD;             // 200k
    const int nNC  = N_NODES * HD;                // 6.4M
    const int nEH  = ETOT * NHEAD;                // 3.4M
    const int nEQ  = ETOT * (HD / 4);             // 27.2M (edge x 4-ch groups)

    // ping-pong: layer0 in=x -> bufA ; layer1 in=bufA -> bufB ; layer2 in=bufB -> bufA
    const float* cur = x;
    float* nxt = bufA;

    for (int l = 0; l < NLAYERS; ++l) {
        const float* Wl  = Wall  + (size_t)l * IN_DIM * HD;
        const float* asl = a_src + (size_t)l * NHEAD * DHEAD;
        const float* adl = a_dst + (size_t)l * NHEAD * DHEAD;
        const float* bl  = bias  + (size_t)l * HD;

        // H = cur @ W  (WMMA f32, LDS-staged A tile)
        gemm_wmma_kernel<<<N_NODES / 16, 256, 0, stream>>>(cur, Wl, hbuf);

        // attention logits
        attn_scores_kernel<<<(nNH + B - 1) / B, B, 0, stream>>>(hbuf, asl, adl, als, ald);

        // init accumulators
        fill_kernel<<<(nNH + B - 1) / B, B, 0, stream>>>(mmax, -3.402823466e38f, nNH);
        fill_kernel<<<(nNH + B - 1) / B, B, 0, stream>>>(denom, 0.f, nNH);
        fill_kernel<<<(nNC + B - 1) / B, B, 0, stream>>>(nxt, 0.f, nNC);

        // softmax over in-edges (max, exp+sum), then weighted aggregation
        edge_max_kernel<<<(nEH + B - 1) / B, B, 0, stream>>>(ei, als, ald, escr, mmax);
        edge_exp_kernel<<<(nEH + B - 1) / B, B, 0, stream>>>(ei, mmax, escr, denom);
        edge_agg_kernel<<<(nEQ + B - 1) / B, B, 0, stream>>>(ei, escr, denom, hbuf, nxt);

        // bias + encoder LeakyReLU(0.2), in place
        bias_leaky_kernel<<<(nNC + B - 1) / B, B, 0, stream>>>(nxt, bl, nxt);

        // rotate ping-pong
        cur = nxt;
        nxt = (nxt == bufA) ? bufB : bufA;
    }

    // global add pool  (final features live in `cur`)
    fill_kernel<<<(G_GRAPHS * HD + B - 1) / B, B, 0, stream>>>(pooled, 0.f, G_GRAPHS * HD);
    pool_kernel<<<(nNC + B - 1) / B, B, 0, stream>>>(cur, batch, pooled);

    // batch-norm stats (one block) + fused BN-apply/FC
    bn_stats_kernel<<<1, 256, 0, stream>>>(pooled, cond, gamma, beta, scale, shift);
    fc_kernel<<<(G_GRAPHS * LATD + B - 1) / B, B, 0, stream>>>(
        pooled, cond, scale, shift, fcw, fcb, out);
}